// NonLocalBlock_60988535603408
// MI455X (gfx1250) — compile-verified
//
#include <hip/hip_runtime.h>
#include <math.h>

// ---------------------------------------------------------------------------
// NonLocalBlock on MI455X (gfx1250, wave32, WMMA + TDM).
//
// Reference collapses algebraically:
//   softmax over size-1 axis == 1  -> theta/phi branches dead
//   y2[b,i,h,w] = g_x[b,w]         -> W-conv -> S[o]*g_x[b,w] + w_b[o]
//   BN(train)                      -> out = x + a[o]*(g_x[b,w]-mu) + beta[o]
// with g_x[b,o] = max_{h,w}( sum_i g_w[o,i]*x[b,i,h,w] ) + g_b[o]
//      S[o]    = sum_i w_w[o,i]
//      a[o]    = S[o]*gamma[o]*rsqrt(S[o]^2*var_g + eps)
// ---------------------------------------------------------------------------

#define CC      192
#define NPIX    (CC * CC)        // 36864 pixels per (b, channel) plane
#define TILE_N  64               // pixels per block tile
#define CHUNKS  (NPIX / TILE_N)  // 576
#define LDS_STRIDE 72            // padded row stride (floats); TDM pad: 64 DW + 8 DW
#define BN_EPS  1e-5f

typedef __attribute__((ext_vector_type(2))) float        v2f;
typedef __attribute__((ext_vector_type(8))) float        v8f;
typedef __attribute__((ext_vector_type(4))) unsigned int v4u;
typedef __attribute__((ext_vector_type(8))) int          v8i;
typedef __attribute__((ext_vector_type(4))) int          v4i;

#if defined(__HIP_DEVICE_COMPILE__) && __has_builtin(__builtin_amdgcn_wmma_f32_16x16x4_f32)
static __device__ __forceinline__ v8f wmma_f32x4(v2f a, v2f b, v8f c) {
  // D = A(16x4 f32) x B(4x16 f32) + C(16x16 f32)
  return __builtin_amdgcn_wmma_f32_16x16x4_f32(false, a, false, b, (short)0, c,
                                               false, false);
}
#else
static __device__ __forceinline__ v8f wmma_f32x4(v2f a, v2f b, v8f c) {
  c[0] += a.x * b.x + a.y * b.y;   // keep-compiling fallback (histogram exposes it)
  return c;
}
#endif

#if defined(__HIP_DEVICE_COMPILE__) && __has_builtin(__builtin_amdgcn_tensor_load_to_lds)
#define HAVE_TDM 1
#else
#define HAVE_TDM 0
#endif

static __device__ __forceinline__ void wait_tensorcnt0() {
#if defined(__HIP_DEVICE_COMPILE__)
#if __has_builtin(__builtin_amdgcn_s_wait_tensorcnt)
  __builtin_amdgcn_s_wait_tensorcnt(0);
#else
  asm volatile("s_wait_tensorcnt 0" ::: "memory");
#endif
#endif
}

// ---------------------------------------------------------------------------
// Kernel 1: g-conv (1x1 conv == GEMM  W(192x192) * X(192xN)) fused with a
// per-row max over this block's 64-pixel tile.  Output: partial[b][chunk][o].
// Grid: B*CHUNKS blocks x 256 threads (8 waves).
// Wave w: column group (w&3)*16 .. +15,  M-tiles (w>>2)*6 .. +5.
// X tile staged into LDS by the Tensor Data Mover (one DMA per block) with
// hardware row padding 64+8 DWORDs -> bank-conflict-free stride of 72 floats.
// ---------------------------------------------------------------------------
__global__ __launch_bounds__(256) void gconv_max_kernel(
    const float* __restrict__ x,     // (B, C, NPIX)
    const float* __restrict__ gw,    // (C, C) row-major [o][i]
    float* __restrict__ partial)     // (B, CHUNKS, C)
{
  __shared__ float lx[CC * LDS_STRIDE];  // [k][p], p in [0,64), stride 72
  __shared__ float smax[4][CC];          // per-column-group row maxima

  const int chunk = blockIdx.x % CHUNKS;
  const int b     = blockIdx.x / CHUNKS;
  const int n0    = chunk * TILE_N;
  const float* xb = x + (size_t)b * CC * NPIX;
  const int t     = threadIdx.x;

#if HAVE_TDM
  // --- Tensor Data Mover: 2D tile 64 x 192 (row stride NPIX) -> LDS with
  // hardware padding (pad_interval=5 -> every 64 DWORDs, pad_amount=7 -> +8 DW).
  if (t < 32) {  // TDM ignores EXEC; exactly one wave issues the DMA
    const unsigned long long gaddr = (unsigned long long)(const void*)(xb + n0);
    const unsigned lds_base = (unsigned)(size_t)&lx[0];  // flat->LDS: addr[31:0]

    v4u g0;
    g0[0] = 1u;                                        // count=1 (valid user D#)
    g0[1] = lds_base;                                  // lds_addr (bytes)
    g0[2] = (unsigned)(gaddr & 0xFFFFFFFFu);           // global_addr[31:0]
    g0[3] = (unsigned)((gaddr >> 32) & 0x01FFFFFFu)    // global_addr[56:32]
          | (2u << 30);                                // type=2 ("image")

    v8i g1;
    g1[0] = (2 << 16)        // data_size = 4 bytes
          | (1 << 20)        // pad_enable
          | (5 << 22)        // pad_interval: 64 DWORDs
          | (7 << 25);       // pad_amount : 8 DWORDs  -> LDS row stride 72 floats
    g1[1] = (int)((NPIX & 0xFFFF) << 16);              // tensor_dim0[15:0]
    g1[2] = (int)((NPIX >> 16) & 0xFFFF)               // tensor_dim0[31:16]
          | (int)((CC & 0xFFFF) << 16);                // tensor_dim1[15:0]
    g1[3] = (int)((TILE_N & 0xFFFF) << 16);            // tensor_dim1 hi=0 | tile_dim0=64
    g1[4] = (int)(CC & 0xFFFF);                        // tile_dim1=192, tile_dim2=0
    g1[5] = (int)NPIX;                                 // tensor_dim0_stride[31:0]
    g1[6] = 0;                                         // stride hi / dim1_stride lo
    g1[7] = 0;

    const v4i zz4 = (v4i){0, 0, 0, 0};
    const v8i zz8 = (v8i){0, 0, 0, 0, 0, 0, 0, 0};
    __builtin_amdgcn_tensor_load_to_lds(g0, g1, zz4, zz4, zz8, 0);
    wait_tensorcnt0();
  }
  __syncthreads();
#else
  // Fallback staging: 3072 float4, 12 per thread, coalesced.
#pragma unroll
  for (int it = 0; it < 12; ++it) {
    int f4 = it * 256 + t;
    int k  = f4 >> 4;
    int p4 = f4 & 15;
    float4 v = *reinterpret_cast<const float4*>(xb + (size_t)k * NPIX + n0 + p4 * 4);
    float* dst = &lx[k * LDS_STRIDE + p4 * 4];
    dst[0] = v.x; dst[1] = v.y; dst[2] = v.z; dst[3] = v.w;
  }
  __syncthreads();
#endif

  const int wave = t >> 5;
  const int lane = t & 31;
  const int nsub = wave & 3;            // 16-pixel column group
  const int mgrp = wave >> 2;           // 0: rows 0..95, 1: rows 96..191
  const int col  = nsub * 16 + (lane & 15);
  const int koff = (lane >> 4) * 2;     // half-wave K offset (A/B frag layout)
  const int mrow = lane & 15;           // row within 16x16 M-tile (A frag)

  v8f acc[6];
#pragma unroll
  for (int i = 0; i < 6; ++i) acc[i] = (v8f){0.f, 0.f, 0.f, 0.f, 0.f, 0.f, 0.f, 0.f};

  // Weight-row pointers for this lane's 6 M-tiles (k = koff baked in).
  const float* wr[6];
#pragma unroll
  for (int tt = 0; tt < 6; ++tt)
    wr[tt] = gw + (size_t)((mgrp * 6 + tt) * 16 + mrow) * CC + koff;

  // Software-pipelined K loop (48 steps of K=4): prefetch step kk+1's
  // fragments before issuing step kk's 6 WMMAs, so the per-iteration
  // s_wait_loadcnt overlaps with matrix work instead of serializing it.
  v2f bcur, bnxt = (v2f){0.f, 0.f};
  v2f acur[6], anxt[6];
  bcur.x = lx[koff * LDS_STRIDE + col];
  bcur.y = lx[(koff + 1) * LDS_STRIDE + col];
#pragma unroll
  for (int tt = 0; tt < 6; ++tt) {
    acur[tt].x = wr[tt][0];
    acur[tt].y = wr[tt][1];
    anxt[tt]   = (v2f){0.f, 0.f};
  }

#pragma unroll 4
  for (int kk = 0; kk < CC / 4; ++kk) {
    if (kk < CC / 4 - 1) {
      const int kn = (kk + 1) * 4 + koff;
      bnxt.x = lx[kn * LDS_STRIDE + col];
      bnxt.y = lx[(kn + 1) * LDS_STRIDE + col];
#pragma unroll
      for (int tt = 0; tt < 6; ++tt) {
        anxt[tt].x = wr[tt][(kk + 1) * 4];
        anxt[tt].y = wr[tt][(kk + 1) * 4 + 1];
      }
    }
#pragma unroll
    for (int tt = 0; tt < 6; ++tt)
      acc[tt] = wmma_f32x4(acur[tt], bcur, acc[tt]);
    bcur = bnxt;
#pragma unroll
    for (int tt = 0; tt < 6; ++tt) acur[tt] = anxt[tt];
  }

  // C/D layout: VGPR j holds M = tile_m0 + j (lanes 0-15) / + 8 + j (lanes 16-31),
  // N = n0 + (lane&15).  Max across the 16 lanes of each half-wave per row.
#pragma unroll
  for (int tt = 0; tt < 6; ++tt) {
#pragma unroll
    for (int j = 0; j < 8; ++j) {
      float v = acc[tt][j];
#pragma unroll
      for (int xm = 1; xm <= 8; xm <<= 1)
        v = fmaxf(v, __shfl_xor(v, xm, 32));   // masks < 16: stays in half-wave
      const int row = (mgrp * 6 + tt) * 16 + j + 8 * (lane >> 4);
      if ((lane & 15) == 0) smax[nsub][row] = v;
    }
  }
  __syncthreads();

  if (t < CC) {
    float v = fmaxf(fmaxf(smax[0][t], smax[1][t]),
                    fmaxf(smax[2][t], smax[3][t]));
    partial[((size_t)b * CHUNKS + chunk) * CC + t] = v;
  }
}

// ---------------------------------------------------------------------------
// Kernel 2: reduce partial maxima over chunks, add g bias.
// ---------------------------------------------------------------------------
__global__ __launch_bounds__(256) void reduce_max_kernel(
    const float* __restrict__ partial,  // (B, CHUNKS, C)
    const float* __restrict__ gb,       // (C)
    float* __restrict__ g_x)            // (B, C)
{
  const int o = blockIdx.x % CC;
  const int b = blockIdx.x / CC;
  float m = -INFINITY;
  for (int c = threadIdx.x; c < CHUNKS; c += 256)
    m = fmaxf(m, partial[((size_t)b * CHUNKS + c) * CC + o]);

  __shared__ float red[256];
  red[threadIdx.x] = m;
  __syncthreads();
  for (int s = 128; s > 0; s >>= 1) {
    if (threadIdx.x < s) red[threadIdx.x] = fmaxf(red[threadIdx.x], red[threadIdx.x + s]);
    __syncthreads();
  }
  if (threadIdx.x == 0) g_x[b * CC + o] = red[0] + gb[o];
}

// ---------------------------------------------------------------------------
// Kernel 3: BN stats + per-channel scale.  One block, 256 threads.
// ---------------------------------------------------------------------------
__global__ __launch_bounds__(256) void stats_kernel(
    const float* __restrict__ g_x,    // (B, C) = 768 values
    const float* __restrict__ ww,     // (C, C)
    const float* __restrict__ gamma,  // (C)
    float* __restrict__ a_out,        // (C)
    float* __restrict__ mu_out)       // (1)
{
  __shared__ float s_sum[256], s_sq[256];
  float ls = 0.f, lq = 0.f;
  for (int i = threadIdx.x; i < 4 * CC; i += 256) {
    float v = g_x[i];
    ls += v;
    lq += v * v;
  }
  s_sum[threadIdx.x] = ls;
  s_sq[threadIdx.x]  = lq;
  __syncthreads();
  for (int s = 128; s > 0; s >>= 1) {
    if (threadIdx.x < s) {
      s_sum[threadIdx.x] += s_sum[threadIdx.x + s];
      s_sq[threadIdx.x]  += s_sq[threadIdx.x + s];
    }
    __syncthreads();
  }
  const float inv_n = 1.0f / (4.0f * CC);
  const float mu  = s_sum[0] * inv_n;
  const float var = s_sq[0] * inv_n - mu * mu;   // biased, matches jnp.var
  if (threadIdx.x == 0) *mu_out = mu;

  if (threadIdx.x < CC) {
    const int o = threadIdx.x;
    float S = 0.f;
    for (int i = 0; i < CC; ++i) S += ww[(size_t)o * CC + i];
    a_out[o] = S * gamma[o] * rsqrtf(S * S * var + BN_EPS);
  }
}

// ---------------------------------------------------------------------------
// Kernel 4: out[b,o,h,w] = x[b,o,h,w] + a[o]*(g_x[b,w]-mu) + beta[o]
// ---------------------------------------------------------------------------
__global__ __launch_bounds__(256) void epilogue_kernel(
    const float* __restrict__ x,
    const float* __restrict__ g_x,    // (B, C)
    const float* __restrict__ a,      // (C)
    const float* __restrict__ beta,   // (C)
    const float* __restrict__ mu_p,
    float* __restrict__ out)
{
  const size_t idx = (size_t)blockIdx.x * 256 + threadIdx.x;  // float4 index
  const int    w4  = (int)(idx % 48);
  const size_t r2  = (idx / 48) / CC;   // (b, o)
  const int    o   = (int)(r2 % CC);
  const int    b   = (int)(r2 / CC);

  const float  mu = *mu_p;
  const float  ao = a[o];
  const float  bo = beta[o];
  const float4 xv = reinterpret_cast<const float4*>(x)[idx];
  const float4 gv = *reinterpret_cast<const float4*>(g_x + b * CC + w4 * 4);

  float4 ov;
  ov.x = xv.x + ao * (gv.x - mu) + bo;
  ov.y = xv.y + ao * (gv.y - mu) + bo;
  ov.z = xv.z + ao * (gv.z - mu) + bo;
  ov.w = xv.w + ao * (gv.w - mu) + bo;
  reinterpret_cast<float4*>(out)[idx] = ov;
}

// ---------------------------------------------------------------------------
extern "C" void kernel_launch(void* const* d_in, const int* in_sizes, int n_in,
                              void* d_out, int out_size, void* d_ws, size_t ws_size,
                              hipStream_t stream) {
  const float* x     = (const float*)d_in[0];
  const float* g_w   = (const float*)d_in[1];
  const float* g_b   = (const float*)d_in[2];
  // d_in[3..6]: theta_w/b, phi_w/b dead (softmax over size-1 axis == 1).
  const float* w_w   = (const float*)d_in[7];
  // d_in[8]: w_b cancels exactly in training-mode batch norm.
  const float* gamma = (const float*)d_in[9];
  const float* beta  = (const float*)d_in[10];
  float* out = (float*)d_out;

  // Workspace layout (floats); every element is rewritten each call.
  float* ws      = (float*)d_ws;
  float* partial = ws;                              // 4*576*192 = 442368
  float* g_x     = ws + (size_t)4 * CHUNKS * CC;    // 768
  float* a       = g_x + 4 * CC;                    // 192
  float* mu      = a + CC;                          // 1

  gconv_max_kernel<<<4 * CHUNKS, 256, 0, stream>>>(x, g_w, partial);
  reduce_max_kernel<<<4 * CC, 256, 0, stream>>>(partial, g_b, g_x);
  stats_kernel<<<1, 256, 0, stream>>>(g_x, w_w, gamma, a, mu);

  const int total4 = 4 * CC * CC * CC / 4;          // 7,077,888 float4
  epilogue_kernel<<<total4 / 256, 256, 0, stream>>>(x, g_x, a, beta, mu, out);
}